// ScaledDotProductAttention_19593640804536
// MI455X (gfx1250) — compile-verified
//
#include <hip/hip_runtime.h>
#include <math.h>

typedef float v2f __attribute__((ext_vector_type(2)));
typedef float v8f __attribute__((ext_vector_type(8)));

constexpr int   B = 32;
constexpr int   S = 8192;
constexpr int   D = 64;
constexpr float SCALE = 0.125f;            // 1 / temperature, temperature = sqrt(64)

constexpr int WAVES          = 8;          // waves per block (256 threads, wave32)
constexpr int KEYS_PER_WAVE  = 64;         // 4 WMMA tiles of 16 keys
constexpr int CHUNKS         = S / (WAVES * KEYS_PER_WAVE);   // 16 blocks per batch

// workspace layout (floats)
constexpr int NPART     = B * CHUNKS;      // 512 partials
constexpr int WS_O_OFF  = 0;               // [NPART][D]
constexpr int WS_M_OFF  = NPART * D;       // [NPART]
constexpr int WS_L_OFF  = WS_M_OFF + NPART;// [NPART]
constexpr int WS_ML_OFF = WS_L_OFF + NPART;// [B][2]  final (M, L) per batch

// ---------------------------------------------------------------------------
// Pass 1: flash-decode partials. One block = one (batch, S-chunk). Each wave
// owns 64 keys: scores via V_WMMA_F32_16X16X4_F32 (q replicated across the 16
// B-matrix columns so the D=64 dot-product reduction happens inside the matrix
// pipe), online softmax, coalesced float2 V accumulation.
// ---------------------------------------------------------------------------
__global__ __launch_bounds__(WAVES * 32)
void attn_partial_kernel(const float* __restrict__ q,
                         const float* __restrict__ k,
                         const float* __restrict__ v,
                         float* __restrict__ attn,   // [B][S] raw scaled scores (pass 1)
                         float* __restrict__ ws)
{
    const int tid   = threadIdx.x;
    const int lane  = tid & 31;
    const int wave  = tid >> 5;
    const int chunk = blockIdx.x;           // 0..CHUNKS-1
    const int b     = blockIdx.y;           // 0..B-1

    __shared__ __align__(16) float qs[D];
    __shared__ float sm[WAVES];
    __shared__ float sl[WAVES];
    __shared__ __align__(16) float so[WAVES][D];

    if (tid < D) qs[tid] = q[b * D + tid];
    __syncthreads();

    const int waveKeyBase = (chunk * WAVES + wave) * KEYS_PER_WAVE;
    const int colBase     = (lane >> 4) << 1;   // lanes 0-15: K=0,1 ; lanes 16-31: K=2,3

    float m = -INFINITY, l = 0.f, o0 = 0.f, o1 = 0.f;

    for (int t = 0; t < KEYS_PER_WAVE / 16; ++t) {
        const int s0     = waveKeyBase + t * 16;
        const int keyRow = s0 + (lane & 15);
        const float* kp  = k + ((size_t)b * S + keyRow) * D;

        // --- QK^T for 16 keys: 16 accumulated f32 WMMAs over the D dimension
        v8f acc = {};
        #pragma unroll
        for (int c = 0; c < 16; ++c) {
            v2f a2 = *(const v2f*)(kp + c * 4 + colBase);   // A: 16x4 f32 tile of K
            v2f b2 = *(const v2f*)(qs + c * 4 + colBase);   // B: q chunk, all 16 cols
            acc = __builtin_amdgcn_wmma_f32_16x16x4_f32(
                false, a2, false, b2, (short)0, acc, false, false);
        }

        float ps[8];
        #pragma unroll
        for (int r = 0; r < 8; ++r) ps[r] = acc[r] * SCALE;

        // raw scaled scores -> attn buffer (lane 0: keys s0..s0+7, lane 16: +8..+15)
        if ((lane & 15) == 0) {
            float* ap = attn + (size_t)b * S + s0 + (lane >> 4) * 8;
            *(float4*)(ap)     = make_float4(ps[0], ps[1], ps[2], ps[3]);
            *(float4*)(ap + 4) = make_float4(ps[4], ps[5], ps[6], ps[7]);
        }

        // --- online softmax update
        float tmax = ps[0];
        #pragma unroll
        for (int r = 1; r < 8; ++r) tmax = fmaxf(tmax, ps[r]);
        tmax = fmaxf(tmax, __shfl_xor(tmax, 16, 32));

        const float newm = fmaxf(m, tmax);
        const float resc = __expf(m - newm);
        l *= resc; o0 *= resc; o1 *= resc;
        m = newm;

        // --- accumulate o += p * V ; lane owns dims {2*lane, 2*lane+1}
        const float* vbase = v + (size_t)b * S * D + lane * 2;
        #pragma unroll
        for (int r = 0; r < 8; ++r) {
            const float p   = __expf(ps[r] - m);
            const float plo = __shfl(p, 0, 32);    // key s0+r
            const float phi = __shfl(p, 16, 32);   // key s0+8+r
            l += plo + phi;
            const float2 vlo = *(const float2*)(vbase + (size_t)(s0 + r) * D);
            const float2 vhi = *(const float2*)(vbase + (size_t)(s0 + 8 + r) * D);
            o0 += plo * vlo.x + phi * vhi.x;
            o1 += plo * vlo.y + phi * vhi.y;
        }
    }

    // --- combine the block's 8 waves in LDS, emit one partial per (b, chunk)
    so[wave][lane * 2]     = o0;
    so[wave][lane * 2 + 1] = o1;
    if (lane == 0) { sm[wave] = m; sl[wave] = l; }
    __syncthreads();

    if (tid < D) {
        float M = sm[0];
        #pragma unroll
        for (int i = 1; i < WAVES; ++i) M = fmaxf(M, sm[i]);
        float L = 0.f, O = 0.f;
        #pragma unroll
        for (int i = 0; i < WAVES; ++i) {
            const float f = __expf(sm[i] - M);
            L += f * sl[i];
            O += f * so[i][tid];
        }
        const int idx = b * CHUNKS + chunk;
        ws[WS_O_OFF + idx * D + tid] = O;
        if (tid == 0) { ws[WS_M_OFF + idx] = M; ws[WS_L_OFF + idx] = L; }
    }
}

// ---------------------------------------------------------------------------
// Pass 2: combine CHUNKS partials per batch -> out[b][d], and final (M, L).
// ---------------------------------------------------------------------------
__global__ __launch_bounds__(64)
void attn_combine_kernel(float* __restrict__ out, float* __restrict__ ws)
{
    const int b = blockIdx.x;
    const int t = threadIdx.x;              // 0..63 = output dim
    __shared__ float cm[CHUNKS], cl[CHUNKS];
    if (t < CHUNKS) {
        cm[t] = ws[WS_M_OFF + b * CHUNKS + t];
        cl[t] = ws[WS_L_OFF + b * CHUNKS + t];
    }
    __syncthreads();

    float M = cm[0];
    #pragma unroll
    for (int i = 1; i < CHUNKS; ++i) M = fmaxf(M, cm[i]);
    float L = 0.f, O = 0.f;
    #pragma unroll
    for (int i = 0; i < CHUNKS; ++i) {
        const float f = __expf(cm[i] - M);
        L += f * cl[i];
        O += f * ws[WS_O_OFF + (b * CHUNKS + i) * D + t];
    }
    out[b * D + t] = O / L;
    if (t == 0) { ws[WS_ML_OFF + b * 2] = M; ws[WS_ML_OFF + b * 2 + 1] = L; }
}

// ---------------------------------------------------------------------------
// Pass 3: rewrite raw scores as softmax probabilities: exp(s - M) / L.
// ---------------------------------------------------------------------------
__global__ __launch_bounds__(256)
void attn_normalize_kernel(float* __restrict__ attn, const float* __restrict__ ws)
{
    const int b = blockIdx.y;
    const int s = blockIdx.x * 256 + threadIdx.x;
    const float M = ws[WS_ML_OFF + b * 2];
    const float L = ws[WS_ML_OFF + b * 2 + 1];
    const size_t i = (size_t)b * S + s;
    attn[i] = __expf(attn[i] - M) / L;
}

extern "C" void kernel_launch(void* const* d_in, const int* in_sizes, int n_in,
                              void* d_out, int out_size, void* d_ws, size_t ws_size,
                              hipStream_t stream)
{
    (void)in_sizes; (void)n_in; (void)out_size; (void)ws_size;
    const float* q = (const float*)d_in[0];
    const float* k = (const float*)d_in[1];
    const float* v = (const float*)d_in[2];
    float* out  = (float*)d_out;            // [B][1][D]
    float* attn = out + B * D;              // [B][1][S]
    float* ws   = (float*)d_ws;             // ~134 KB used

    attn_partial_kernel<<<dim3(CHUNKS, B), WAVES * 32, 0, stream>>>(q, k, v, attn, ws);
    attn_combine_kernel<<<dim3(B), 64, 0, stream>>>(out, ws);
    attn_normalize_kernel<<<dim3(S / 256, B), 256, 0, stream>>>(attn, ws);
}